// FastfoodProjection_85847806312577
// MI455X (gfx1250) — compile-verified
//
#include <hip/hip_runtime.h>

typedef __attribute__((ext_vector_type(2))) float v2f;
typedef __attribute__((ext_vector_type(8))) float v8f;

#define IN_FEATURES 4096
#define PADDED      8192
#define THREADS     128   // 4 wave32 per block

// H16[m][k] = (-1)^popcount(m & k)  (symmetric Hadamard, unnormalized)
__device__ __forceinline__ float hsign(int m, int k) {
  return (__popc(m & k) & 1) ? -1.0f : 1.0f;
}

// One radix-16 FWHT stage along the axis of stride S = 1<<LOG, in place in LDS.
// Each wave owns 8 disjoint 16x16 tiles => only the leading barrier is needed.
// A-fragment (H16) layout, 32-bit A 16x4: lanes 0-15 M=0..15, VGPR0 K={0|2}, VGPR1 K={1|3}.
// B-fragment, 32-bit B 4x16: N = lane&15 striped across lanes, VGPR0 K={0|2}, VGPR1 K={1|3}.
// C/D 16x16: N = lane&15, VGPR v holds M = v + 8*(lane>>4).
template<int LOG>
__device__ __forceinline__ void stage16(float* buf, int lane, int wave, const v2f* aH) {
  __syncthreads();
  const int n   = lane & 15;
  const int sel = lane >> 4;
  for (int t = 0; t < 8; ++t) {
    const int tile = wave * 8 + t;
    const int c    = tile * 16 + n;                 // column handled by this lane
    const int lo   = c & ((1 << LOG) - 1);
    const int hi   = c >> LOG;
    const int base = (hi << (LOG + 4)) + lo;        // element index of (c, j=0)
    v8f acc = {0.f, 0.f, 0.f, 0.f, 0.f, 0.f, 0.f, 0.f};
#pragma unroll
    for (int kc = 0; kc < 4; ++kc) {
      const int j0 = kc * 4 + 2 * sel;
      v2f b;
      b.x = buf[base + ((j0    ) << LOG)];
      b.y = buf[base + ((j0 + 1) << LOG)];
      // D = H16 x B + acc   (fp32 exact products, ±1 coefficients)
      acc = __builtin_amdgcn_wmma_f32_16x16x4_f32(false, aH[kc], false, b,
                                                  (short)0, acc, false, false);
    }
#pragma unroll
    for (int v = 0; v < 8; ++v)
      buf[base + ((v + sel * 8) << LOG)] = acc[v];  // M = v + 8*sel, same column c
  }
}

// Unnormalized FWHT of length 8192 = H2 (x) H16 (x) H16 (x) H16, in place in LDS.
__device__ __forceinline__ void fwht8192(float* buf, int tid, int lane, int wave,
                                         const v2f* aH) {
  stage16<0>(buf, lane, wave, aH);   // stride 1
  stage16<4>(buf, lane, wave, aH);   // stride 16
  stage16<8>(buf, lane, wave, aH);   // stride 256
  __syncthreads();
  // radix-2 stage, stride 4096, vectorized as float4 (1024 float4-pairs)
  float4* b4 = (float4*)buf;
#pragma unroll
  for (int i = 0; i < 8; ++i) {
    const int p = tid + i * THREADS;
    float4 a = b4[p];
    float4 b = b4[p + 1024];
    float4 s, d;
    s.x = a.x + b.x; s.y = a.y + b.y; s.z = a.z + b.z; s.w = a.w + b.w;
    d.x = a.x - b.x; d.y = a.y - b.y; d.z = a.z - b.z; d.w = a.w - b.w;
    b4[p] = s;
    b4[p + 1024] = d;
  }
  __syncthreads();
}

__global__ __launch_bounds__(THREADS)
void fastfood_kernel(const float* __restrict__ x, const float* __restrict__ Bv,
                     const float* __restrict__ Gv, const int* __restrict__ perm,
                     float* __restrict__ out) {
  __shared__ float d0[PADDED];
  __shared__ float d1[PADDED];

  const int tid  = threadIdx.x;
  const int lane = tid & 31;
  const int wave = tid >> 5;
  const int row  = blockIdx.x;

  // ---- Stage 0: load x*B (padded with zeros) into LDS, float4 vectorized ----
  const float4* x4 = (const float4*)(x + (size_t)row * IN_FEATURES);
  const float4* B4 = (const float4*)Bv;
  float4* d04 = (float4*)d0;
#pragma unroll
  for (int i = 0; i < 8; ++i) {            // 1024 float4 = 4096 floats
    const int p = tid + i * THREADS;
    float4 xv = x4[p];
    float4 bv = B4[p];
    float4 r;
    r.x = xv.x * bv.x; r.y = xv.y * bv.y; r.z = xv.z * bv.z; r.w = xv.w * bv.w;
    d04[p] = r;
  }
  const float4 z = {0.f, 0.f, 0.f, 0.f};
#pragma unroll
  for (int i = 0; i < 8; ++i)              // zero-pad upper half
    d04[1024 + tid + i * THREADS] = z;

  // ---- Per-lane H16 A-fragments (constant across all stages/tiles) ----
  const int m   = lane & 15;
  const int sel = lane >> 4;
  v2f aH[4];
#pragma unroll
  for (int kc = 0; kc < 4; ++kc) {
    const int k0 = kc * 4 + 2 * sel;
    aH[kc].x = hsign(m, k0);
    aH[kc].y = hsign(m, k0 + 1);
  }

  // ---- FWHT #1 ----
  fwht8192(d0, tid, lane, wave, aH);

  // ---- Permute + G (gather from d0 into d1) ----
#pragma unroll
  for (int i = 0; i < 64; ++i) {
    const int k = tid + i * THREADS;
    d1[k] = d0[perm[k]] * Gv[k];
  }
  __syncthreads();

  // ---- FWHT #2 ----
  fwht8192(d1, tid, lane, wave, aH);

  // ---- Scale by 1/8192 (two 1/sqrt(8192) normalizations folded) and store ----
  const float s = 1.0f / 8192.0f;
  float4* o4  = (float4*)(out + (size_t)row * PADDED);
  const float4* d14 = (const float4*)d1;
#pragma unroll
  for (int i = 0; i < 16; ++i) {           // 2048 float4 = 8192 floats
    const int p = tid + i * THREADS;
    float4 v = d14[p];
    v.x *= s; v.y *= s; v.z *= s; v.w *= s;
    o4[p] = v;
  }
}

extern "C" void kernel_launch(void* const* d_in, const int* in_sizes, int n_in,
                              void* d_out, int out_size, void* d_ws, size_t ws_size,
                              hipStream_t stream) {
  const float* x    = (const float*)d_in[0];
  const float* Bv   = (const float*)d_in[1];
  const float* Gv   = (const float*)d_in[2];
  const int*   perm = (const int*)d_in[3];
  float* out = (float*)d_out;

  const int rows = in_sizes[0] / IN_FEATURES;   // 8192
  hipLaunchKernelGGL(fastfood_kernel, dim3(rows), dim3(THREADS), 0, stream,
                     x, Bv, Gv, perm, out);
}